// HarmonicSynth_47081431499061
// MI455X (gfx1250) — compile-verified
//
#include <hip/hip_runtime.h>
#include <hip/hip_bf16.h>
#include <math.h>

// ---------------------------------------------------------------------------
// HarmonicSynth for MI455X (gfx1250, wave32)
//
// Roofline: outputs ~216 MB -> ~9.3 us at 23.3 TB/s. The upsample(x256) +
// 63-tap Hamming smoothing composes into a coarse-domain banded operator of
// bandwidth <= 4 (63 fine taps span < 1 coarse frame):
//     out[t0+i, c] = sum_{q<4} W[tile][i][q] * coarse[c, jbase(tile)+q]
// so each 16t x 16c tile is exactly ONE V_WMMA_F32_16X16X4_F32:
//     A (16x4) = composed weights W  (per-tile constant, shared by all waves)
//     B (4x16) = raw coarse alphas   (two dword loads per lane)
// W (incl. all reflect/clamp edge effects) is precomputed into a 1.6 MB table
// that lives in L2. Amplitude smoothing reuses W as a 4-tap dot product.
// The fused kernel is then write-bandwidth bound; the 52M sines use explicit
// range reduction to keep v_sin_f32 in-domain.
// ---------------------------------------------------------------------------

#define T_FINE   102400
#define T_COARSE 400
#define NH       64
#define WIN      63
#define HALF     31
#define FACTORF  256.0f
#define NTILE    (T_FINE / 16)     // 6400

typedef float v2f __attribute__((ext_vector_type(2)));
typedef float v8f __attribute__((ext_vector_type(8)));

// ---------------------------------------------------------------------------
// Kernel 1: upsample pitch, /SR, inclusive cumsum per batch (phase in revs).
// ---------------------------------------------------------------------------
__global__ __launch_bounds__(1024)
void hs_scan_kernel(const float* __restrict__ pitch,
                    float* __restrict__ pitch_up,
                    float* __restrict__ csum) {
  __shared__ float lds[1024];
  const int b = blockIdx.x;
  const int tid = threadIdx.x;
  float carry = 0.0f;

  for (int base = 0; base < T_FINE; base += 1024) {
    const int t = base + tid;
    float pos = ((float)t + 0.5f) * (1.0f / FACTORF) - 0.5f;
    pos = fminf(fmaxf(pos, 0.0f), (float)(T_COARSE - 1));
    int lo = (int)pos;
    int hi = lo + 1; if (hi > T_COARSE - 1) hi = T_COARSE - 1;
    const float w = pos - (float)lo;
    const float p = (pitch[b * T_COARSE + lo] * (1.0f - w) +
                     pitch[b * T_COARSE + hi] * w) * (1.0f / 16000.0f);
    pitch_up[(size_t)b * T_FINE + t] = p;

    lds[tid] = p;
    __syncthreads();
    for (int off = 1; off < 1024; off <<= 1) {
      const float add = (tid >= off) ? lds[tid - off] : 0.0f;
      __syncthreads();
      lds[tid] += add;
      __syncthreads();
    }
    csum[(size_t)b * T_FINE + t] = lds[tid] + carry;
    carry += lds[1023];
    __syncthreads();
  }
}

// ---------------------------------------------------------------------------
// Kernel 2: build the composed upsample+smooth weight table.
// Wtab[tile][i][q] = sum_k win[k] * (interp weight of coarse jbase+q at
// reflect-padded fine index t0+i+k). Also stores jbase per tile.
// One thread per (tile, i, q): 6400*64 threads, 63-MAC loop each.
// ---------------------------------------------------------------------------
__global__ __launch_bounds__(256)
void hs_wtab_kernel(const float* __restrict__ win,
                    float* __restrict__ Wtab,
                    int* __restrict__ jb) {
  const int gid = blockIdx.x * 256 + threadIdx.x;
  if (gid >= NTILE * 64) return;
  const int tile = gid >> 6;
  const int e = gid & 63;
  const int i = e >> 2;
  const int q = e & 3;
  const int t0 = tile * 16;

  int smin = t0 - HALF; if (smin < 0) smin = 0;
  float posmin = ((float)smin + 0.5f) * (1.0f / FACTORF) - 0.5f;
  posmin = fminf(fmaxf(posmin, 0.0f), (float)(T_COARSE - 1));
  int jbase = (int)posmin; if (jbase > T_COARSE - 4) jbase = T_COARSE - 4;
  if (e == 0) jb[tile] = jbase;

  const int jq = jbase + q;
  float acc = 0.0f;
  for (int k = 0; k < WIN; ++k) {
    int s = t0 + i + k - HALF;
    s = (s < 0) ? -s : s;                        // reflect at 0
    if (s >= T_FINE) s = 2 * (T_FINE - 1) - s;   // reflect at top
    float pos = ((float)s + 0.5f) * (1.0f / FACTORF) - 0.5f;
    pos = fminf(fmaxf(pos, 0.0f), (float)(T_COARSE - 1));
    const int lo = (int)pos;
    int hi = lo + 1; if (hi > T_COARSE - 1) hi = T_COARSE - 1;
    const float w = pos - (float)lo;
    float contrib = 0.0f;
    if (lo == jq) contrib += 1.0f - w;
    if (hi == jq) contrib += w;
    acc += win[k] * contrib;
  }
  Wtab[gid] = acc;
}

// ---------------------------------------------------------------------------
// Kernel 3: fused smoothing (1 WMMA per wave) + amplitude smoothing (4-tap) +
// mask/normalize + sine bank. Grid (NTILE, 8 batches), 128 threads = 4 waves.
// ---------------------------------------------------------------------------
__global__ __launch_bounds__(128)
void hs_fused_kernel(const float* __restrict__ amp_in,
                     const float* __restrict__ alphas_in,
                     const float* __restrict__ pitch_up,
                     const float* __restrict__ csum,
                     const float* __restrict__ Wtab,
                     const int* __restrict__ jb,
                     float* __restrict__ y_out,
                     float* __restrict__ amp_out,
                     float* __restrict__ al_out) {
  __shared__ float sm[NH][17];    // smoothed alphas tile (pad vs bank conflict)
  __shared__ float redA[8][16];
  __shared__ float redY[8][16];
  __shared__ float ampS[16];
  __shared__ float phS[16];
  __shared__ float ptS[16];
  __shared__ float invS[16];

  const int tid = threadIdx.x;
  const int b = blockIdx.y;
  const int tile = blockIdx.x;
  const int t0 = tile * 16;
  const int jbase = jb[tile];

  const int wave = tid >> 5;
  const int lane = tid & 31;
  const int n = lane & 15;                 // A row M; B/D col N
  const int kbase = (lane < 16) ? 0 : 2;   // K halves split across lane halves
  const int c = wave * 16 + n;

  // A operand: composed weights W[i=n][kbase..kbase+1] (per-tile, L2-hot)
  const float* wp = Wtab + (size_t)tile * 64 + n * 4 + kbase;
  v2f Av; Av[0] = wp[0]; Av[1] = wp[1];

  // B operand: raw coarse alphas, B[k,n] = coarse[c, jbase+k]
  const float* ap = alphas_in + ((size_t)b * NH + c) * T_COARSE + jbase + kbase;
  v2f Bv; Bv[0] = ap[0]; Bv[1] = ap[1];

  v8f acc = {};
  acc = __builtin_amdgcn_wmma_f32_16x16x4_f32(
      /*neg_a=*/false, Av, /*neg_b=*/false, Bv,
      /*c_mod=*/(short)0, acc, /*reuse_a=*/false, /*reuse_b=*/false);

  // D layout: VGPR r -> M=r (lanes 0-15) / M=r+8 (lanes 16-31); N = lane&15
  const int mbase = (lane < 16) ? 0 : 8;
#pragma unroll
  for (int r = 0; r < 8; ++r) sm[c][mbase + r] = acc[r];

  // ---- amplitude smoothing (same composed 4-tap) + per-time scalars ----
  if (tid < 16) {
    const int t = t0 + tid;
    phS[tid] = csum[(size_t)b * T_FINE + t];
    ptS[tid] = pitch_up[(size_t)b * T_FINE + t];
    const float* mp = amp_in + (size_t)b * T_COARSE + jbase;
    const float* wr = Wtab + (size_t)tile * 64 + tid * 4;
    ampS[tid] = wr[0] * mp[0] + wr[1] * mp[1] + wr[2] * mp[2] + wr[3] * mp[3];
  }
  __syncthreads();

  // ---- mask, normalize, sine bank ----
  const int i = tid & 15;      // time within tile
  const int g = tid >> 4;      // harmonic group of 8
  const float ph = phS[i];
  const float pt = ptS[i];

  float pm[8];
  float psA = 0.0f, psY = 0.0f;
#pragma unroll
  for (int q = 0; q < 8; ++q) {
    const int ch = g * 8 + q;
    const float hm = (float)(ch + 1);
    const float mask = (pt * hm < 0.5f) ? 1.0f : 0.0f;
    const float mv = sm[ch][i] * mask;
    pm[q] = mv;
    psA += mv;
    float rev = hm * ph;
    rev -= floorf(rev);                       // keep v_sin_f32 in-range
    psY += __sinf(6.283185307179586f * rev) * mv;
  }
  redA[g][i] = psA;
  redY[g][i] = psY;
  __syncthreads();

  if (tid < 16) {
    float ta = 0.0f, ty = 0.0f;
#pragma unroll
    for (int gg = 0; gg < 8; ++gg) { ta += redA[gg][tid]; ty += redY[gg][tid]; }
    const float inv = 1.0f / ta;
    invS[tid] = inv;
    const size_t o = (size_t)b * T_FINE + t0 + tid;
    y_out[o] = ampS[tid] * ty * inv;
    amp_out[o] = ampS[tid];
  }
  __syncthreads();

  const float inv = invS[i];
#pragma unroll
  for (int q = 0; q < 8; ++q) {
    const int ch = g * 8 + q;
    al_out[((size_t)b * NH + ch) * T_FINE + t0 + i] = pm[q] * inv;
  }
}

// ---------------------------------------------------------------------------
extern "C" void kernel_launch(void* const* d_in, const int* in_sizes, int n_in,
                              void* d_out, int out_size, void* d_ws, size_t ws_size,
                              hipStream_t stream) {
  const float* amp    = (const float*)d_in[0];  // [8,1,400]
  const float* alphas = (const float*)d_in[1];  // [8,64,400]
  const float* pitch  = (const float*)d_in[2];  // [8,1,400]
  const float* win    = (const float*)d_in[3];  // [63]

  float* ws = (float*)d_ws;
  float* pitch_up = ws;                              // [8, 102400]
  float* csum     = pitch_up + (size_t)8 * T_FINE;   // [8, 102400]
  float* Wtab     = csum + (size_t)8 * T_FINE;       // [6400, 16, 4]
  int*   jbtab    = (int*)(Wtab + (size_t)NTILE * 64);  // [6400]

  float* out     = (float*)d_out;
  float* y_out   = out;                                    // y        [8,1,T]
  float* amp_out = out + (size_t)8 * T_FINE;               // amplitude[8,1,T]
  float* al_out  = out + (size_t)16 * T_FINE;              // alphas  [8,64,T]

  hs_scan_kernel<<<8, 1024, 0, stream>>>(pitch, pitch_up, csum);
  hs_wtab_kernel<<<(NTILE * 64 + 255) / 256, 256, 0, stream>>>(win, Wtab, jbtab);
  hs_fused_kernel<<<dim3(NTILE, 8), 128, 0, stream>>>(
      amp, alphas, pitch_up, csum, Wtab, jbtab, y_out, amp_out, al_out);
}